// DifferentiableXMap_46265387712687
// MI455X (gfx1250) — compile-verified
//
#include <hip/hip_runtime.h>

typedef __attribute__((ext_vector_type(2))) float v2f;
typedef __attribute__((ext_vector_type(8))) float v8f;

#define GRID_N 96
#define NVOX   (GRID_N * GRID_N * GRID_N)   // 884736
#define G2     (GRID_N * GRID_N)

// Trilinear border-clamped sample of both batch volumes at wrapped coords.
__device__ __forceinline__ void sample_pair(
    float tc0, float tc1, float tc2,
    const float* __restrict__ vol0, const float* __restrict__ vol1,
    float& acc0, float& acc1)
{
    const float G   = (float)GRID_N;
    const float Gm1 = (float)(GRID_N - 1);
    // Python-style mod: x - floor(x/G)*G  -> [0, G)
    float x = tc0 - floorf(tc0 * (1.0f / G)) * G;
    float y = tc1 - floorf(tc1 * (1.0f / G)) * G;
    float z = tc2 - floorf(tc2 * (1.0f / G)) * G;
    // border padding == clamp to [0, G-1]
    x = fminf(fmaxf(x, 0.0f), Gm1);
    y = fminf(fmaxf(y, 0.0f), Gm1);
    z = fminf(fmaxf(z, 0.0f), Gm1);
    float xf = floorf(x), yf = floorf(y), zf = floorf(z);
    int x0 = (int)xf, y0 = (int)yf, z0 = (int)zf;
    int x1 = min(x0 + 1, GRID_N - 1);
    int y1 = min(y0 + 1, GRID_N - 1);
    int z1 = min(z0 + 1, GRID_N - 1);
    float fx = x - xf, fy = y - yf, fz = z - zf;

    int zy00 = z0 * G2 + y0 * GRID_N;
    int zy01 = z0 * G2 + y1 * GRID_N;
    int zy10 = z1 * G2 + y0 * GRID_N;
    int zy11 = z1 * G2 + y1 * GRID_N;

    {   // batch 0
        float c000 = vol0[zy00 + x0], c001 = vol0[zy00 + x1];
        float c010 = vol0[zy01 + x0], c011 = vol0[zy01 + x1];
        float c100 = vol0[zy10 + x0], c101 = vol0[zy10 + x1];
        float c110 = vol0[zy11 + x0], c111 = vol0[zy11 + x1];
        float c00 = c000 + fx * (c001 - c000);
        float c01 = c010 + fx * (c011 - c010);
        float c10 = c100 + fx * (c101 - c100);
        float c11 = c110 + fx * (c111 - c110);
        float c0  = c00  + fy * (c01  - c00);
        float c1  = c10  + fy * (c11  - c10);
        acc0 += c0 + fz * (c1 - c0);
    }
    {   // batch 1 (same addresses, second volume)
        float c000 = vol1[zy00 + x0], c001 = vol1[zy00 + x1];
        float c010 = vol1[zy01 + x0], c011 = vol1[zy01 + x1];
        float c100 = vol1[zy10 + x0], c101 = vol1[zy10 + x1];
        float c110 = vol1[zy11 + x0], c111 = vol1[zy11 + x1];
        float c00 = c000 + fx * (c001 - c000);
        float c01 = c010 + fx * (c011 - c010);
        float c10 = c100 + fx * (c101 - c100);
        float c11 = c110 + fx * (c111 - c110);
        float c0  = c00  + fy * (c01  - c00);
        float c1  = c10  + fy * (c11  - c10);
        acc1 += c0 + fz * (c1 - c0);
    }
}

// One wave32 handles 16 consecutive voxels (same i,j; k = k0..k0+15).
// Coordinates for all 8 ops come from two V_WMMA_F32_16X16X4_F32:
//   D(32x16, padded 4 rows/op) = [R | t*96] (32x4) x [i, j, k, 1]^T (4x16)
// WMMA D layout then hands each lane 4 complete (op, voxel) coordinate
// triples in its own VGPRs -- no cross-lane shuffles needed.
__global__ __launch_bounds__(256) void xmap_wmma_kernel(
    const float* __restrict__ density,
    const float* __restrict__ Rm,     // (8,3,3)
    const float* __restrict__ tv,     // (8,3)
    const float* __restrict__ off,    // (3,)
    float* __restrict__ out)          // (2,96,96,96)
{
    const int lane  = threadIdx.x & 31;
    const int wave  = (blockIdx.x << 3) | (threadIdx.x >> 5); // 8 waves/block
    const int vbase = wave << 4;                              // 16 voxels/wave
    const int col   = lane & 15;
    const bool hi   = lane >= 16;

    const int k0 = vbase % GRID_N;
    const int j  = (vbase / GRID_N) % GRID_N;
    const int i  = vbase / G2;

    // Warm the cachelines holding this wave's direct density terms
    // (emits gfx1250 global_prefetch_b8; loaded again at the epilogue).
    __builtin_prefetch(density + (hi ? NVOX : 0) + vbase + col, 0, 3);

    const float o0 = off[0], o1 = off[1], o2 = off[2];

    // ---- A operands: rows m = op*4 + comp (comp 3 = zero pad) -------------
    // A 16x4 layout: lanes 0-15 -> K=0,1 in (v0,v1); lanes 16-31 -> K=2,3.
    const int   m    = lane & 15;
    const int   comp = m & 3;
    const int   cc   = comp < 3 ? comp : 2;       // clamp pad row's index
    const float msk  = (comp < 3) ? 1.0f : 0.0f;  // zero the pad row
    const int   opLo = m >> 2;                    // ops 0..3 (first WMMA)
    const int   opHi = opLo + 4;                  // ops 4..7 (second WMMA)

    v2f aLo, aHi;
    if (!hi) {  // K = 0,1 -> R columns 0,1
        aLo.x = msk * Rm[opLo * 9 + cc * 3 + 0];
        aLo.y = msk * Rm[opLo * 9 + cc * 3 + 1];
        aHi.x = msk * Rm[opHi * 9 + cc * 3 + 0];
        aHi.y = msk * Rm[opHi * 9 + cc * 3 + 1];
    } else {    // K = 2,3 -> R column 2, translation*GRID (homogeneous)
        aLo.x = msk * Rm[opLo * 9 + cc * 3 + 2];
        aLo.y = msk * tv[opLo * 3 + cc] * (float)GRID_N;
        aHi.x = msk * Rm[opHi * 9 + cc * 3 + 2];
        aHi.y = msk * tv[opHi * 3 + cc] * (float)GRID_N;
    }

    // ---- B operand: homogeneous voxel coords (i, j, k, 1) for 16 columns --
    // B 4x16 layout: lanes 0-15 -> K=0,1 in (v0,v1); lanes 16-31 -> K=2,3.
    v2f b;
    b.x = hi ? ((float)(k0 + col) + o2) : ((float)i + o0);
    b.y = hi ? 1.0f                     : ((float)j + o1);

    v8f czero = {0.f, 0.f, 0.f, 0.f, 0.f, 0.f, 0.f, 0.f};
    v8f d1 = __builtin_amdgcn_wmma_f32_16x16x4_f32(
        false, aLo, false, b, (short)0, czero, false, false);
    v8f d2 = __builtin_amdgcn_wmma_f32_16x16x4_f32(
        false, aHi, false, b, (short)0, czero, false, false);

    // D layout: lane (lo half) holds rows 0-7 -> ops {opBase, opBase+1}
    // with coords at v[0..2] and v[4..6]; hi half holds rows 8-15.
    const float* vol0 = density;
    const float* vol1 = density + NVOX;
    float acc0 = 0.0f, acc1 = 0.0f;

    sample_pair(d1[0], d1[1], d1[2], vol0, vol1, acc0, acc1);
    sample_pair(d1[4], d1[5], d1[6], vol0, vol1, acc0, acc1);
    sample_pair(d2[0], d2[1], d2[2], vol0, vol1, acc0, acc1);
    sample_pair(d2[4], d2[5], d2[6], vol0, vol1, acc0, acc1);

    // Each voxel's 8 ops are split across lane and lane^16: fold halves.
    acc0 += __shfl_xor(acc0, 16, 32);
    acc1 += __shfl_xor(acc1, 16, 32);

    if (!hi) {
        const int v = vbase + col;
        out[v]        = (density[v]        + acc0) * 0.125f;
        out[NVOX + v] = (density[NVOX + v] + acc1) * 0.125f;
    }
}

extern "C" void kernel_launch(void* const* d_in, const int* in_sizes, int n_in,
                              void* d_out, int out_size, void* d_ws, size_t ws_size,
                              hipStream_t stream) {
    (void)in_sizes; (void)n_in; (void)out_size; (void)d_ws; (void)ws_size;
    const float* density = (const float*)d_in[0];
    const float* Rm      = (const float*)d_in[1];
    const float* tv      = (const float*)d_in[2];
    const float* off     = (const float*)d_in[3];
    float*       out     = (float*)d_out;

    dim3 block(256);                       // 8 wave32s
    dim3 grid(NVOX / 128);                 // 16 voxels per wave -> 6912 blocks
    hipLaunchKernelGGL(xmap_wmma_kernel, grid, block, 0, stream,
                       density, Rm, tv, off, out);
}